// GAT_12893491822857
// MI455X (gfx1250) — compile-verified
//
#include <hip/hip_runtime.h>

#define N_IN   128   // IN_DIM
#define N_OF   64    // HEADS*HID
#define HEADS  4
#define HID    16
#define NEG    0.2f

typedef float v2f __attribute__((ext_vector_type(2)));
typedef float v8f __attribute__((ext_vector_type(8)));

// ---------- order-preserving float <-> uint mapping for atomicMax ----------
__device__ __forceinline__ unsigned fmap(float x) {
    unsigned u = __float_as_uint(x);
    return (u & 0x80000000u) ? ~u : (u | 0x80000000u);
}
__device__ __forceinline__ float funmap(unsigned u) {
    return __uint_as_float((u & 0x80000000u) ? (u & 0x7FFFFFFFu) : ~u);
}

// ---------- 0: init scratch (emax=mapped(-inf), denom=0, agg=0) ----------
__global__ void gat_init(unsigned* __restrict__ emax, float* __restrict__ denom,
                         float* __restrict__ agg, int n_nodes) {
    int t = blockIdx.x * blockDim.x + threadIdx.x;
    int nAgg = n_nodes * N_OF;
    if (t < nAgg) agg[t] = 0.0f;
    if (t < n_nodes * HEADS) { emax[t] = 0x007FFFFFu; denom[t] = 0.0f; }
}

// ---------- 1: projection GEMM via V_WMMA_F32_16X16X4_F32 ----------
// One wave computes a 16x64 output tile (4 accumulators). W staged in LDS.
__global__ void __launch_bounds__(256)
gat_proj_wmma(const float* __restrict__ feats, const float* __restrict__ W,
              float* __restrict__ feat_out, int n_tiles) {
    __shared__ float Wlds[N_IN * N_OF]; // 32 KB
    for (int i = threadIdx.x; i < N_IN * N_OF; i += blockDim.x) Wlds[i] = W[i];
    __syncthreads();

    const int lane  = threadIdx.x & 31;
    const int wave  = threadIdx.x >> 5;
    const int tile  = blockIdx.x * (blockDim.x >> 5) + wave;
    if (tile >= n_tiles) return;           // wave-uniform exit: EXEC stays all-ones

    const int rlo   = lane & 15;           // row-within-tile (A) / column (B,D)
    const int khalf = lane >> 4;           // selects K pair {0,1} vs {2,3}
    const float* __restrict__ arow = feats + (size_t)(tile * 16 + rlo) * N_IN;

    v8f acc0 = {}, acc1 = {}, acc2 = {}, acc3 = {};
    for (int k = 0; k < N_IN; k += 4) {
        const int ka = k + khalf * 2;
        v2f a; a.x = arow[ka]; a.y = arow[ka + 1];
        const float* wr0 = Wlds + ka * N_OF;
        const float* wr1 = wr0 + N_OF;
        v2f b0 = { wr0[rlo +  0], wr1[rlo +  0] };
        v2f b1 = { wr0[rlo + 16], wr1[rlo + 16] };
        v2f b2 = { wr0[rlo + 32], wr1[rlo + 32] };
        v2f b3 = { wr0[rlo + 48], wr1[rlo + 48] };
        acc0 = __builtin_amdgcn_wmma_f32_16x16x4_f32(false, a, false, b0, (short)0, acc0, false, false);
        acc1 = __builtin_amdgcn_wmma_f32_16x16x4_f32(false, a, false, b1, (short)0, acc1, false, false);
        acc2 = __builtin_amdgcn_wmma_f32_16x16x4_f32(false, a, false, b2, (short)0, acc2, false, false);
        acc3 = __builtin_amdgcn_wmma_f32_16x16x4_f32(false, a, false, b3, (short)0, acc3, false, false);
    }
    // D layout: lane = column N (mod 16), VGPR r = row M=r (lanes 0-15) / M=r+8 (16-31)
    const int mbase = khalf * 8;
    for (int r = 0; r < 8; ++r) {
        float* orow = feat_out + (size_t)(tile * 16 + mbase + r) * N_OF;
        orow[rlo +  0] = acc0[r];
        orow[rlo + 16] = acc1[r];
        orow[rlo + 32] = acc2[r];
        orow[rlo + 48] = acc3[r];
    }
}

// ---------- 2: el/er per (node, head) ----------
__global__ void gat_elr(const float* __restrict__ feat, const float* __restrict__ attn_l,
                        const float* __restrict__ attn_r, float* __restrict__ el,
                        float* __restrict__ er, int n_nodes) {
    int t = blockIdx.x * blockDim.x + threadIdx.x;   // t = n*HEADS + h
    if (t >= n_nodes * HEADS) return;
    int h = t & (HEADS - 1), n = t >> 2;
    const float* f  = feat + (size_t)n * N_OF + h * HID;
    const float* al = attn_l + h * HID;
    const float* ar = attn_r + h * HID;
    float sl = 0.f, sr = 0.f;
#pragma unroll
    for (int d = 0; d < HID; ++d) { float v = f[d]; sl += v * al[d]; sr += v * ar[d]; }
    el[t] = sl; er[t] = sr;
}

__device__ __forceinline__ float edge_logit(const float* el, const float* er,
                                            int s, int d, int h) {
    float x = el[s * HEADS + h] + er[d * HEADS + h];
    return x > 0.f ? x : NEG * x;
}

// ---------- 3: segment max over dst ----------
__global__ void gat_edge_max(const int* __restrict__ src, const int* __restrict__ dst,
                             const float* __restrict__ el, const float* __restrict__ er,
                             unsigned* __restrict__ emax, int n_edges) {
    int t = blockIdx.x * blockDim.x + threadIdx.x;   // t = e*HEADS + h
    if (t >= n_edges * HEADS) return;
    int h = t & (HEADS - 1), e = t >> 2;
    int s = src[e], d = dst[e];
    atomicMax(&emax[d * HEADS + h], fmap(edge_logit(el, er, s, d, h)));
}

// ---------- 4: segment sum of exp ----------
__global__ void gat_edge_denom(const int* __restrict__ src, const int* __restrict__ dst,
                               const float* __restrict__ el, const float* __restrict__ er,
                               const unsigned* __restrict__ emax, float* __restrict__ denom,
                               int n_edges) {
    int t = blockIdx.x * blockDim.x + threadIdx.x;
    if (t >= n_edges * HEADS) return;
    int h = t & (HEADS - 1), e = t >> 2;
    int s = src[e], d = dst[e];
    float x = edge_logit(el, er, s, d, h);
    float m = funmap(emax[d * HEADS + h]);
    atomicAdd(&denom[d * HEADS + h], __expf(x - m));
}

// ---------- 5: weighted scatter-add aggregation ----------
__global__ void gat_edge_agg(const int* __restrict__ src, const int* __restrict__ dst,
                             const float* __restrict__ el, const float* __restrict__ er,
                             const unsigned* __restrict__ emax, const float* __restrict__ denom,
                             const float* __restrict__ feat, float* __restrict__ agg,
                             int n_edges) {
    int t = blockIdx.x * blockDim.x + threadIdx.x;   // t = e*HEADS + h
    if (t >= n_edges * HEADS) return;
    int h = t & (HEADS - 1), e = t >> 2;
    int s = src[e], d = dst[e];
    float x = edge_logit(el, er, s, d, h);
    float m = funmap(emax[d * HEADS + h]);
    float alpha = __expf(x - m) / denom[d * HEADS + h];
    const float4* fs = (const float4*)(feat + (size_t)s * N_OF + h * HID);
    float* ad = agg + (size_t)d * N_OF + h * HID;
#pragma unroll
    for (int q = 0; q < 4; ++q) {
        float4 v = fs[q];
        atomicAdd(&ad[q * 4 + 0], alpha * v.x);
        atomicAdd(&ad[q * 4 + 1], alpha * v.y);
        atomicAdd(&ad[q * 4 + 2], alpha * v.z);
        atomicAdd(&ad[q * 4 + 3], alpha * v.w);
    }
}

// ---------- 6: head mean + bias ----------
__global__ void gat_finalize(const float* __restrict__ agg, const float* __restrict__ bias,
                             float* __restrict__ out, int n_nodes) {
    int t = blockIdx.x * blockDim.x + threadIdx.x;   // t = n*HID + dcol
    if (t >= n_nodes * HID) return;
    int dcol = t & (HID - 1), n = t >> 4;
    const float* a = agg + (size_t)n * N_OF;
    float sum = 0.f;
#pragma unroll
    for (int h = 0; h < HEADS; ++h) sum += a[h * HID + dcol] + bias[h * HID + dcol];
    out[t] = sum * 0.25f;
}

extern "C" void kernel_launch(void* const* d_in, const int* in_sizes, int n_in,
                              void* d_out, int out_size, void* d_ws, size_t ws_size,
                              hipStream_t stream) {
    const float* features = (const float*)d_in[0];
    const float* W        = (const float*)d_in[1];
    const float* attn_l   = (const float*)d_in[2];
    const float* attn_r   = (const float*)d_in[3];
    const float* bias     = (const float*)d_in[4];
    const int*   src      = (const int*)d_in[5];
    const int*   dst      = (const int*)d_in[6];
    float*       out      = (float*)d_out;

    const int n_nodes = in_sizes[0] / N_IN;
    const int n_edges = in_sizes[5];

    // workspace layout (floats)
    float* ws    = (float*)d_ws;
    float*    feat  = ws;                                  // n_nodes*64
    float*    el    = feat  + (size_t)n_nodes * N_OF;      // n_nodes*4
    float*    er    = el    + (size_t)n_nodes * HEADS;     // n_nodes*4
    unsigned* emax  = (unsigned*)(er + (size_t)n_nodes * HEADS); // n_nodes*4
    float*    denom = (float*)emax + (size_t)n_nodes * HEADS;    // n_nodes*4
    float*    agg   = denom + (size_t)n_nodes * HEADS;     // n_nodes*64

    const int TPB = 256;

    int initN = n_nodes * N_OF;
    gat_init<<<(initN + TPB - 1) / TPB, TPB, 0, stream>>>(emax, denom, agg, n_nodes);

    int n_tiles = (n_nodes + 15) / 16;                     // 3125, exact
    int wavesPerBlock = TPB / 32;
    gat_proj_wmma<<<(n_tiles + wavesPerBlock - 1) / wavesPerBlock, TPB, 0, stream>>>(
        features, W, feat, n_tiles);

    int nh = n_nodes * HEADS;
    gat_elr<<<(nh + TPB - 1) / TPB, TPB, 0, stream>>>(feat, attn_l, attn_r, el, er, n_nodes);

    int eh = n_edges * HEADS;
    gat_edge_max<<<(eh + TPB - 1) / TPB, TPB, 0, stream>>>(src, dst, el, er, emax, n_edges);
    gat_edge_denom<<<(eh + TPB - 1) / TPB, TPB, 0, stream>>>(src, dst, el, er, emax, denom, n_edges);
    gat_edge_agg<<<(eh + TPB - 1) / TPB, TPB, 0, stream>>>(src, dst, el, er, emax, denom,
                                                           feat, agg, n_edges);

    int on = n_nodes * HID;
    gat_finalize<<<(on + TPB - 1) / TPB, TPB, 0, stream>>>(agg, bias, out, n_nodes);
}